// DHN_21809843929922
// MI455X (gfx1250) — compile-verified
//
#include <hip/hip_runtime.h>
#include <hip/hip_bf16.h>
#include <math.h>

// DHN pairwise-hashing loss for MI455X (gfx1250, wave32).
//   sim = 0.5 * b @ b^T           -> fp32 WMMA (V_WMMA_F32_16X16X4_F32)
//   loss1: per-row softplus over (pos, neg) pairs, LDS-staged, deterministic
//   loss2: mean((b - sign(b))^2)
// Outputs: d_out[0]=total, d_out[1]=loss1, d_out[2]=loss2 (fp32)

#define DHN_N    2048
#define DHN_BITS 64
#define DHN_ALPHA 5.0f

typedef __attribute__((ext_vector_type(2))) float v2f;
typedef __attribute__((ext_vector_type(8))) float v8f;

// ---------------------------------------------------------------------------
// Kernel 1: sim[i][j] = 0.5 * dot(b[i], b[j])  via f32 WMMA, one wave / tile.
// A 16x4 f32 layout: lane<16 -> M=lane, K={k0,k0+1}; lane>=16 -> M=lane-16,
// K={k0+2,k0+3}. B 4x16 mirrors with N in place of M; B = b^T so both sides
// load two consecutive K elements of a row of b.
// ---------------------------------------------------------------------------
__global__ __launch_bounds__(32) void dhn_sim_wmma(const float* __restrict__ b,
                                                   float* __restrict__ sim) {
    const int lane = threadIdx.x;      // 0..31, wave32
    const int half = lane >> 4;        // 0 or 1
    const int lm   = lane & 15;
    const int rowBase = blockIdx.y * 16;
    const int colBase = blockIdx.x * 16;

    const float* arow = b + (size_t)(rowBase + lm) * DHN_BITS + half * 2;
    const float* brow = b + (size_t)(colBase + lm) * DHN_BITS + half * 2;

    v8f c = {};
#pragma unroll
    for (int k0 = 0; k0 < DHN_BITS; k0 += 4) {
        v2f av, bv;
        av.x = arow[k0];
        av.y = arow[k0 + 1];
        bv.x = brow[k0];
        bv.y = brow[k0 + 1];
        // 8 args: (neg_a, A, neg_b, B, c_mod, C, reuse_a, reuse_b)
        c = __builtin_amdgcn_wmma_f32_16x16x4_f32(
                false, av, false, bv, (short)0, c, false, false);
    }

    // C/D layout: VGPR r -> M = r + 8*half, N = lm
    float* out = sim + (size_t)(rowBase + half * 8) * DHN_N + (colBase + lm);
#pragma unroll
    for (int r = 0; r < 8; ++r) {
        out[(size_t)r * DHN_N] = 0.5f * c[r];
    }
}

// ---------------------------------------------------------------------------
// Kernel 2: one block per row i.
//   pos = { j : y[j]==y[i] } (includes i itself, matching the reference)
//   row_loss = sum_{p in pos, n in neg} softplus(-(th_p - th_n - alpha)) / npairs
// Deterministic: positives compacted by a serial scan, fixed reduction tree.
// ---------------------------------------------------------------------------
__global__ __launch_bounds__(256) void dhn_row_loss(const float* __restrict__ sim,
                                                    const int* __restrict__ y,
                                                    float* __restrict__ rowLoss,
                                                    float* __restrict__ rowValid) {
    __shared__ int   ys[DHN_N];        // 8 KB
    __shared__ int   posIdx[DHN_N];    // 8 KB (worst case: all positive)
    __shared__ float thetaPos[DHN_N];  // 8 KB
    __shared__ float red[256];
    __shared__ int   nPosSh;

    const int i   = blockIdx.x;
    const int tid = threadIdx.x;

    for (int j = tid; j < DHN_N; j += 256) ys[j] = y[j];
    __syncthreads();

    const int yi = ys[i];

    // Deterministic compaction of the positive set (fixed order).
    if (tid == 0) {
        int np = 0;
        for (int j = 0; j < DHN_N; ++j)
            if (ys[j] == yi) posIdx[np++] = j;
        nPosSh = np;
    }
    __syncthreads();

    const int nPos = nPosSh;
    const int nNeg = DHN_N - nPos;
    const float* theta = sim + (size_t)i * DHN_N;

    for (int p = tid; p < nPos; p += 256) thetaPos[p] = theta[posIdx[p]];
    __syncthreads();

    // Pair loop: each thread owns a strided set of negatives.
    float acc = 0.0f;
    for (int j = tid; j < DHN_N; j += 256) {
        if (ys[j] != yi) {
            const float base = -theta[j] - DHN_ALPHA;  // t = th_p - th_n - alpha
            for (int p = 0; p < nPos; ++p) {
                float t = thetaPos[p] + base;
                t = fminf(fmaxf(t, -100.0f), 50.0f);
                // log1p(exp(t)) - t  == softplus(-t), exactly as the reference
                acc += log1pf(expf(t)) - t;
            }
        }
    }

    red[tid] = acc;
    __syncthreads();
    for (int s = 128; s > 0; s >>= 1) {
        if (tid < s) red[tid] += red[tid + s];
        __syncthreads();
    }

    if (tid == 0) {
        const bool valid = (nPos > 0) && (nNeg > 0);
        const float npairs = (float)max(nPos * nNeg, 1);
        rowLoss[i]  = valid ? red[0] / npairs : 0.0f;
        rowValid[i] = valid ? 1.0f : 0.0f;
    }
}

// ---------------------------------------------------------------------------
// Kernel 3: quantization loss partials: sum((b - sign(b))^2) per block.
// ---------------------------------------------------------------------------
__global__ __launch_bounds__(256) void dhn_quant(const float* __restrict__ b,
                                                 float* __restrict__ partial) {
    __shared__ float red[256];
    const int tid = threadIdx.x;
    float acc = 0.0f;
    const int total = DHN_N * DHN_BITS;
    for (int idx = blockIdx.x * 256 + tid; idx < total; idx += gridDim.x * 256) {
        float x = b[idx];
        float s = (x > 0.0f) ? 1.0f : ((x < 0.0f) ? -1.0f : 0.0f);
        float d = x - s;
        acc += d * d;
    }
    red[tid] = acc;
    __syncthreads();
    for (int s = 128; s > 0; s >>= 1) {
        if (tid < s) red[tid] += red[tid + s];
        __syncthreads();
    }
    if (tid == 0) partial[blockIdx.x] = red[0];
}

// ---------------------------------------------------------------------------
// Kernel 4: deterministic finalize -> (total, loss1, loss2)
// ---------------------------------------------------------------------------
__global__ __launch_bounds__(256) void dhn_finalize(const float* __restrict__ rowLoss,
                                                    const float* __restrict__ rowValid,
                                                    const float* __restrict__ quantPartial,
                                                    float* __restrict__ out) {
    __shared__ float redL[256];
    __shared__ float redC[256];
    const int tid = threadIdx.x;
    float sL = 0.0f, sC = 0.0f;
    for (int i = tid; i < DHN_N; i += 256) {
        sL += rowLoss[i];
        sC += rowValid[i];
    }
    redL[tid] = sL;
    redC[tid] = sC;
    __syncthreads();
    for (int s = 128; s > 0; s >>= 1) {
        if (tid < s) { redL[tid] += redL[tid + s]; redC[tid] += redC[tid + s]; }
        __syncthreads();
    }
    if (tid == 0) {
        float q = 0.0f;
        for (int i = 0; i < 64; ++i) q += quantPartial[i];
        const float loss2 = q / (float)(DHN_N * DHN_BITS);
        const float cnt   = redC[0];
        const float loss1 = (cnt > 0.0f) ? redL[0] / fmaxf(cnt, 1.0f) : 0.0f;
        out[0] = loss1 + loss2;   // LAMBDA == 1.0
        out[1] = loss1;
        out[2] = loss2;
    }
}

// ---------------------------------------------------------------------------
extern "C" void kernel_launch(void* const* d_in, const int* in_sizes, int n_in,
                              void* d_out, int out_size, void* d_ws, size_t ws_size,
                              hipStream_t stream) {
    (void)in_sizes; (void)n_in; (void)out_size; (void)ws_size;

    const float* b = (const float*)d_in[0];   // [2048, 64] fp32
    const int*   y = (const int*)d_in[1];     // [2048] int32
    float*       out = (float*)d_out;         // 3 floats

    float* ws        = (float*)d_ws;
    float* sim       = ws;                                   // 2048*2048 fp32 (16 MB)
    float* rowLoss   = ws + (size_t)DHN_N * DHN_N;           // 2048
    float* rowValid  = rowLoss + DHN_N;                      // 2048
    float* quantPart = rowValid + DHN_N;                     // 64

    dim3 gGemm(DHN_N / 16, DHN_N / 16);                      // 128 x 128 tiles
    dhn_sim_wmma<<<gGemm, 32, 0, stream>>>(b, sim);
    dhn_row_loss<<<DHN_N, 256, 0, stream>>>(sim, y, rowLoss, rowValid);
    dhn_quant<<<64, 256, 0, stream>>>(b, quantPart);
    dhn_finalize<<<1, 256, 0, stream>>>(rowLoss, rowValid, quantPart, out);
}